// NeuralFingerprint_13752485282098
// MI455X (gfx1250) — compile-verified
//
#include <hip/hip_runtime.h>
#include <hip/hip_bf16.h>

#define N_NODES 50000
#define FEAT    128
#define FP_LEN  2048
#define RADIUS  4
#define N_EDGES 600000
#define BN_EPS  1e-5f

typedef float v2f __attribute__((ext_vector_type(2)));
typedef float v8f __attribute__((ext_vector_type(8)));

// CDNA5 fp32 tensor op: D(16x16,f32) = A(16x4,f32) x B(4x16,f32) + C
// A frag (per lane): row = lane&15, k = k0 + 2*(lane>>4) + {0,1}
// B frag (per lane): col = lane&15, k = k0 + 2*(lane>>4) + {0,1}
// C/D (per lane):    vgpr r -> row = 8*(lane>>4) + r, col = lane&15
__device__ __forceinline__ v8f wmma_f32(v2f a, v2f b, v8f c) {
  return __builtin_amdgcn_wmma_f32_16x16x4_f32(false, a, false, b, (short)0, c,
                                               false, false);
}

// ---------------------------------------------------------------- utilities
__global__ void zero_kernel(float* __restrict__ p, int n) {
  int i = blockIdx.x * blockDim.x + threadIdx.x;
  if (i < n) p[i] = 0.0f;
}

__global__ void copy4_kernel(const float4* __restrict__ s, float4* __restrict__ d, int n4) {
  int i = blockIdx.x * blockDim.x + threadIdx.x;
  if (i < n4) d[i] = s[i];
}

// --------------------------------------------------- edge scatter-add (agg)
// 32 threads per edge, 4 consecutive floats per thread (coalesced float4 read)
__global__ void edge_scatter_kernel(const float* __restrict__ feats,
                                    const long long* __restrict__ ei,
                                    float* __restrict__ agg) {
  long long g = (long long)blockIdx.x * blockDim.x + threadIdx.x;
  int e = (int)(g >> 5);
  if (e >= N_EDGES) return;
  int j = ((int)g & 31) * 4;
  int s = (int)ei[e];
  int d = (int)ei[N_EDGES + e];
  float4 v = *(const float4*)(feats + (size_t)s * FEAT + j);
  float* o = agg + (size_t)d * FEAT + j;
  atomicAdd(o + 0, v.x);
  atomicAdd(o + 1, v.y);
  atomicAdd(o + 2, v.z);
  atomicAdd(o + 3, v.w);
}

// ------------------------------------------- hash layer: relu(agg @ W + b)
// one wave per 16-row stripe; 8 N-tiles of 16, K=128 in steps of 4
__global__ void hash_gemm_kernel(const float* __restrict__ A,   // [N][128]
                                 const float* __restrict__ W,   // [128][128]
                                 const float* __restrict__ bias,// [128]
                                 float* __restrict__ H) {       // [N][128]
  const int lane = threadIdx.x & 31;
  const int wave = threadIdx.x >> 5;
  const int mb = blockIdx.x * 8 + wave;
  if (mb >= N_NODES / 16) return;
  const int m0    = mb * 16;
  const int rowA  = m0 + (lane & 15);
  const int khalf = (lane >> 4) * 2;
  const int col   = lane & 15;

  v8f acc[8] = {};
  for (int k0 = 0; k0 < FEAT; k0 += 4) {
    const int kb = k0 + khalf;
    v2f a;
    a.x = A[(size_t)rowA * FEAT + kb];
    a.y = A[(size_t)rowA * FEAT + kb + 1];
#pragma unroll
    for (int t = 0; t < 8; ++t) {
      v2f b;
      b.x = W[(size_t)kb * FEAT + t * 16 + col];
      b.y = W[(size_t)(kb + 1) * FEAT + t * 16 + col];
      acc[t] = wmma_f32(a, b, acc[t]);
    }
  }
  const int rowD0 = (lane >> 4) * 8;
#pragma unroll
  for (int t = 0; t < 8; ++t) {
    const int c  = t * 16 + col;
    const float bb = bias[c];
#pragma unroll
    for (int r = 0; r < 8; ++r) {
      float v = acc[t][r] + bb;
      H[(size_t)(m0 + rowD0 + r) * FEAT + c] = v > 0.0f ? v : 0.0f;
    }
  }
}

// --------------------------------------------------- BatchNorm batch stats
__global__ void bn_stats_kernel(const float* __restrict__ H,
                                float* __restrict__ sums) {  // [256]: sum, sumsq
  const int col  = threadIdx.x & (FEAT - 1);
  const int rsub = threadIdx.x >> 7;  // 0..1
  float s = 0.0f, ss = 0.0f;
  for (int row = blockIdx.x * 2 + rsub; row < N_NODES; row += gridDim.x * 2) {
    float v = H[(size_t)row * FEAT + col];
    s += v;
    ss += v * v;
  }
  atomicAdd(&sums[col], s);
  atomicAdd(&sums[FEAT + col], ss);
}

__global__ void bn_finalize_kernel(const float* __restrict__ sums,
                                   float* __restrict__ stats) {  // [256]: mu, rsigma
  int c = threadIdx.x;
  if (c < FEAT) {
    const float inv_n = 1.0f / (float)N_NODES;
    float mu  = sums[c] * inv_n;
    float var = sums[FEAT + c] * inv_n - mu * mu;  // biased var
    stats[c]        = mu;
    stats[FEAT + c] = rsqrtf(var + BN_EPS);
  }
}

__global__ void bn_apply_kernel(const float* __restrict__ H,
                                const float* __restrict__ stats,
                                const float* __restrict__ gamma,
                                const float* __restrict__ beta,
                                float* __restrict__ out) {
  size_t i = (size_t)blockIdx.x * blockDim.x + threadIdx.x;
  if (i >= (size_t)N_NODES * FEAT) return;
  int c = (int)(i & (FEAT - 1));
  out[i] = (H[i] - stats[c]) * stats[FEAT + c] * gamma[c] + beta[c];
}

// ---------------- fused GEMM + row softmax + column reduce into fingerprint
// block = 16 waves (512 thr), owns 32 rows. Wave w computes cols
// [w*128, w*128+128) for BOTH 16-row tiles -> each B fragment feeds 2 WMMAs
// (halves L2 weight traffic vs 16-row blocks). Logits (32x2048 f32 = 256 KB)
// + A tile (16 KB) live in the 320 KB WGP LDS.
__global__ void softmax_fp_kernel(const float* __restrict__ X,    // [N][128]
                                  const float* __restrict__ W,    // [128][2048]
                                  const float* __restrict__ bias, // [2048]
                                  float* __restrict__ fp) {       // [2048]
  extern __shared__ float smem[];
  float* logits = smem;                 // 32 * 2048
  float* Atile  = smem + 32 * FP_LEN;   // 32 * 128

  const int lane = threadIdx.x & 31;
  const int wave = threadIdx.x >> 5;    // 0..15
  const int m0   = blockIdx.x * 32;
  const int rows_valid = (N_NODES - m0 < 32) ? (N_NODES - m0) : 32;

  // stage the 32x128 A block through LDS (zero-pad tail rows; EXEC stays all-1
  // for WMMA, pad rows are masked out of the final reduction instead)
  for (int i = threadIdx.x; i < 32 * FEAT; i += blockDim.x) {
    int r = i >> 7;
    Atile[i] = (r < rows_valid) ? X[(size_t)m0 * FEAT + i] : 0.0f;
  }
  __syncthreads();

  const int col   = lane & 15;
  const int rowA  = lane & 15;
  const int khalf = (lane >> 4) * 2;
  const int n0    = wave * 128;   // 8 col-tiles of 16

  v8f acc[2][8] = {};
  for (int k0 = 0; k0 < FEAT; k0 += 4) {
    const int kb = k0 + khalf;
    v2f a0, a1;
    a0.x = Atile[rowA * FEAT + kb];
    a0.y = Atile[rowA * FEAT + kb + 1];
    a1.x = Atile[(16 + rowA) * FEAT + kb];
    a1.y = Atile[(16 + rowA) * FEAT + kb + 1];
#pragma unroll
    for (int t = 0; t < 8; ++t) {
      const float* wp = W + (size_t)kb * FP_LEN + n0 + t * 16 + col;
      v2f b;
      b.x = wp[0];
      b.y = wp[FP_LEN];  // row kb+1
      acc[0][t] = wmma_f32(a0, b, acc[0][t]);
      acc[1][t] = wmma_f32(a1, b, acc[1][t]);
    }
  }

  // spill logits (+bias) to LDS
  const int rowD0 = (lane >> 4) * 8;
#pragma unroll
  for (int rt = 0; rt < 2; ++rt) {
#pragma unroll
    for (int t = 0; t < 8; ++t) {
      const int c    = n0 + t * 16 + col;
      const float bb = bias[c];
#pragma unroll
      for (int r = 0; r < 8; ++r)
        logits[(rt * 16 + rowD0 + r) * FP_LEN + c] = acc[rt][t][r] + bb;
    }
  }
  __syncthreads();

  // row softmax: each of the 16 waves owns 2 rows; wave32 shuffle reductions
#pragma unroll
  for (int rr = 0; rr < 2; ++rr) {
    float* lr = logits + (wave * 2 + rr) * FP_LEN;
    float m = -3.402823466e38f;
    for (int c = lane; c < FP_LEN; c += 32) m = fmaxf(m, lr[c]);
#pragma unroll
    for (int off = 16; off > 0; off >>= 1) m = fmaxf(m, __shfl_xor(m, off, 32));
    float s = 0.0f;
    for (int c = lane; c < FP_LEN; c += 32) {
      float e = __expf(lr[c] - m);
      lr[c] = e;
      s += e;
    }
#pragma unroll
    for (int off = 16; off > 0; off >>= 1) s += __shfl_xor(s, off, 32);
    const float inv = 1.0f / s;
    for (int c = lane; c < FP_LEN; c += 32) lr[c] *= inv;
  }
  __syncthreads();

  // column-reduce the valid rows of the block, accumulate into fingerprint
  for (int c = threadIdx.x; c < FP_LEN; c += blockDim.x) {
    float s = 0.0f;
    for (int r = 0; r < rows_valid; ++r) s += logits[r * FP_LEN + c];
    atomicAdd(&fp[c], s);
  }
}

// --------------------------------------------------------------------------
extern "C" void kernel_launch(void* const* d_in, const int* in_sizes, int n_in,
                              void* d_out, int out_size, void* d_ws, size_t ws_size,
                              hipStream_t stream) {
  (void)in_sizes; (void)n_in; (void)out_size; (void)ws_size;

  const float*     atoms  = (const float*)d_in[0];
  const long long* ei     = (const long long*)d_in[1];   // int64 edge_index [2][E]
  const float*     init_w = (const float*)d_in[2];
  const float*     init_b = (const float*)d_in[3];
  const float*     hash_w = (const float*)d_in[4];       // [4][128][128]
  const float*     hash_b = (const float*)d_in[5];       // [4][128]
  const float*     soft_w = (const float*)d_in[6];       // [4][128][2048]
  const float*     soft_b = (const float*)d_in[7];       // [4][2048]
  const float*     gamma  = (const float*)d_in[8];
  const float*     beta   = (const float*)d_in[9];
  float* fp = (float*)d_out;

  const size_t NF = (size_t)N_NODES * FEAT;  // 6.4M floats
  float* ws    = (float*)d_ws;
  float* agg   = ws;            // NF
  float* hbuf  = ws + NF;       // NF
  float* featn = ws + 2 * NF;   // NF
  float* sums  = ws + 3 * NF;   // 256
  float* stats = sums + 256;    // 256

  const int smem_bytes = (32 * FP_LEN + 32 * FEAT) * sizeof(float);  // 272 KB
  const int soft_blocks = (N_NODES + 31) / 32;                       // 1563
  const int mblocks16 = N_NODES / 16;                                // 3125 exact

  zero_kernel<<<(FP_LEN + 255) / 256, 256, 0, stream>>>(fp, FP_LEN);

  // initial soft assignment from raw atom features
  softmax_fp_kernel<<<soft_blocks, 512, smem_bytes, stream>>>(atoms, init_w, init_b, fp);

  const float* feats = atoms;
  for (int l = 0; l < RADIUS; ++l) {
    // agg = feats + scatter_add(feats[src] -> dst)
    copy4_kernel<<<(int)(NF / 4 + 255) / 256, 256, 0, stream>>>(
        (const float4*)feats, (float4*)agg, (int)(NF / 4));
    edge_scatter_kernel<<<(N_EDGES * 32 + 255) / 256, 256, 0, stream>>>(feats, ei, agg);

    // h = relu(agg @ hash_w[l] + hash_b[l])
    hash_gemm_kernel<<<(mblocks16 + 7) / 8, 256, 0, stream>>>(
        agg, hash_w + (size_t)l * FEAT * FEAT, hash_b + (size_t)l * FEAT, hbuf);

    // training-mode BatchNorm (biased var)
    zero_kernel<<<1, 256, 0, stream>>>(sums, 2 * FEAT);
    bn_stats_kernel<<<256, 256, 0, stream>>>(hbuf, sums);
    bn_finalize_kernel<<<1, 128, 0, stream>>>(sums, stats);
    bn_apply_kernel<<<(int)(NF + 255) / 256, 256, 0, stream>>>(
        hbuf, stats, gamma, beta, featn);

    // fp += colsum(softmax(featn @ soft_w[l] + soft_b[l]))
    softmax_fp_kernel<<<soft_blocks, 512, smem_bytes, stream>>>(
        featn, soft_w + (size_t)l * FEAT * FP_LEN, soft_b + (size_t)l * FP_LEN, fp);

    feats = featn;
  }
}